// FieldOfJunctions_22342419874281
// MI455X (gfx1250) — compile-verified
//
#include <hip/hip_runtime.h>

// ---------------- problem constants ----------------
#define HH 256
#define WW 256
#define PS 8
#define STR 4
#define NVC 31
#define HPN 63
#define WPN 63
#define NPATCH (HPN*WPN)
#define RECN 24                     // per-patch record: geom(14) + colors(9) + pad

#define TWO_PI_F 6.28318530717958647692f
#define PI_FC    3.14159265358979323846f
#define INV_PI_F 0.31830988618379067154f
#define ETA_INV  100.0f             // 1/ETA, ETA=0.01
#define TAU_F    0.1f
#define DELTA_F  0.05f

typedef float v2f __attribute__((ext_vector_type(2)));
typedef float v8f __attribute__((ext_vector_type(8)));

__device__ __forceinline__ float mod2pi(float x){
  return x - TWO_PI_F * floorf(x * (1.0f / TWO_PI_F));
}

// (t)^35 by squarings (negative base safe, like XLA integer power)
__device__ __forceinline__ float pow35(float t){
  float t2 = t*t, t4 = t2*t2, t8 = t4*t4, t16 = t8*t8, t32 = t16*t16;
  return t32 * t2 * t;
}

struct Geom {
  float x0, y0;
  float s1,c1,s2,c2,s3,c3,s4,c4;
  float sg13, sg42, g13, g42;
};

__device__ __forceinline__ void make_geom(const float* q, Geom& g){
  float b0 = mod2pi(q[0]);
  float b1 = mod2pi(q[1]);
  float b2 = mod2pi(q[2]);
  float lo = fminf(b0,b1), hi = fmaxf(b0,b1);
  float a1 = fminf(lo, b2);                 // sorted angles
  float a3 = fmaxf(hi, b2);
  float a2 = fmaxf(fminf(hi, b2), lo);
  float a4 = 0.5f*(a1+a3) + ((mod2pi(0.5f*(a1-a3)) >= PI_FC) ? PI_FC : 0.0f);
  float r42 = mod2pi(a2 - a4);
  float r13 = mod2pi(a3 - a1);
  g.sg42 = (r42 < PI_FC) ? 1.0f : -1.0f;
  g.sg13 = (r13 < PI_FC) ? 1.0f : -1.0f;
  g.g42 = pow35(r42*INV_PI_F - 1.0f) * TAU_F;
  g.g13 = pow35(r13*INV_PI_F - 1.0f) * TAU_F;
  g.x0 = q[3]; g.y0 = q[4];
  sincosf(a1, &g.s1, &g.c1);
  sincosf(a2, &g.s2, &g.c2);
  sincosf(a3, &g.s3, &g.c3);
  sincosf(a4, &g.s4, &g.c4);
}

__device__ __forceinline__ void dists_at(const Geom& g, float xv, float yv,
                                         float& d13, float& d42){
  float dx = xv - g.x0, dy = yv - g.y0;
  float h1 = -g.s1*dx + g.c1*dy;
  float h2 = -g.s2*dx + g.c2*dy;
  float h3 = -g.s3*dx + g.c3*dy;
  float h4 = -g.s4*dx + g.c4*dy;
  d13 = g.sg13 * fminf(g.sg13*h1, -g.sg13*h3) + g.g13;
  d42 = g.sg42 * fminf(g.sg42*h4, -g.sg42*h2) + g.g42;
}

__device__ __forceinline__ void wedges_from(float d13, float d42,
                                            float& w0, float& w1, float& w2){
  float h0 = 0.5f + INV_PI_F * atanf(d13 * ETA_INV);
  float hb = 0.5f + INV_PI_F * atanf(d42 * ETA_INV);
  w0 = 1.0f - h0;
  w1 = h0 * (1.0f - hb);
  w2 = h0 * hb;
}

// ---------------------------------------------------------------------------
// Kernel 1: one wave32 per patch. Coordinate-descent over 5 params x 31
// candidates. Per candidate, a single f32 WMMA chain computes
//   C = A(16x64) * B(64x16) with A rows = [imgR,imgG,imgB,1,w0,w1,w2],
//   B cols = [w0,w1,w2]  ->  num(3x3), den(3), Gram(3x3)
// and the SSE loss follows in closed form (no 2nd per-pixel pass).
// ---------------------------------------------------------------------------
__global__ __launch_bounds__(32) void foj_descent(const float* __restrict__ x,
                                                  float* __restrict__ rec){
  __shared__ float ilds[3*64];   // img patch, [c][pixel]
  __shared__ float wlds[3*64];   // wedges,    [k][pixel]

  const int patch = blockIdx.x;
  const int hp = patch / WPN, wp = patch % WPN;
  const int lane = threadIdx.x;
  const int p0 = lane << 1;          // this lane owns pixels p0, p0+1 (same row)
  const int i0 = p0 >> 3;
  const int j0 = p0 & 7;             // even
  const int gy = hp*STR + i0;
  const int gx = wp*STR + j0;

  float pv[3][2];
  #pragma unroll
  for (int c=0;c<3;++c){
    pv[c][0] = x[c*(HH*WW) + gy*WW + gx];
    pv[c][1] = x[c*(HH*WW) + gy*WW + gx + 1];
    ilds[c*64 + p0]     = pv[c][0];
    ilds[c*64 + p0 + 1] = pv[c][1];
  }
  __syncthreads();

  // sum img^2 over patch (wave32 butterfly)
  float img2 = 0.0f;
  #pragma unroll
  for (int c=0;c<3;++c) img2 += pv[c][0]*pv[c][0] + pv[c][1]*pv[c][1];
  #pragma unroll
  for (int off=16; off>=1; off>>=1) img2 += __shfl_xor(img2, off, 32);

  // WMMA operand mapping (32-bit, 16x16x4):
  //  A: lane L -> M=L&15 ; VGPR0/1 hold K-pair {hiK, hiK+1}, hiK = 2*(L>>4)
  //  B: lane L -> N=L&15 ; same K-pair striping
  const int M   = lane & 15;
  const int hiK = (lane >> 4) << 1;
  const float2* i2  = (const float2*)ilds;
  const float2* w2  = (const float2*)wlds;
  float2*       w2s = (float2*)wlds;

  // constant A rows (img + ones), preloaded once for all 16 K-chunks
  float2 aconst[16];
  #pragma unroll
  for (int q=0; q<16; ++q){
    int half = (4*q + hiK) >> 1;
    float2 v;
    if (M < 3)       v = i2[M*32 + half];
    else if (M == 3) v = make_float2(1.0f, 1.0f);
    else             v = make_float2(0.0f, 0.0f);
    aconst[q] = v;
  }

  const float xv0 = -1.0f + (2.0f/7.0f) * (float)j0;
  const float xv1 = xv0 + (2.0f/7.0f);
  const float yv0 = -1.0f + (2.0f/7.0f) * (float)i0;

  const bool aW  = (M >= 4) && (M < 7);     // wedge rows of A
  const int  wrc = aW ? (M - 4) : 0;        // clamped (non-divergent LDS loads)
  const bool bW  = (M < 3);                 // wedge cols of B
  const int  bnc = bW ? M : 0;

  float prm[5] = {0.f,0.f,0.f,0.f,0.f};
  Geom  gfin;
  float colfin[9];

  auto evaluate = [&](const float* q, Geom& g, float* colOut) -> float {
    make_geom(q, g);
    float d13a,d42a,d13b,d42b;
    dists_at(g, xv0, yv0, d13a, d42a);
    dists_at(g, xv1, yv0, d13b, d42b);
    float w0a,w1a,w2a, w0b,w1b,w2b;
    wedges_from(d13a, d42a, w0a, w1a, w2a);
    wedges_from(d13b, d42b, w0b, w1b, w2b);
    w2s[0*32 + lane] = make_float2(w0a, w0b);
    w2s[1*32 + lane] = make_float2(w1a, w1b);
    w2s[2*32 + lane] = make_float2(w2a, w2b);
    __syncthreads();

    v8f acc = {0.f,0.f,0.f,0.f,0.f,0.f,0.f,0.f};
    #pragma unroll
    for (int qk=0; qk<16; ++qk){
      int half = (4*qk + hiK) >> 1;
      float2 aw = w2[wrc*32 + half];
      float2 bw = w2[bnc*32 + half];
      float2 ac = aconst[qk];
      v2f A, B;
      A.x = aW ? aw.x : ac.x;
      A.y = aW ? aw.y : ac.y;
      B.x = bW ? bw.x : 0.0f;
      B.y = bW ? bw.y : 0.0f;
      acc = __builtin_amdgcn_wmma_f32_16x16x4_f32(false, A, false, B,
                                                  (short)0, acc, false, false);
    }
    __syncthreads();

    // C layout: VGPR r, lanes 0-15 -> (M=r, N=lane); lanes 16-31 -> (M=r+8).
    // All values we need live in rows 0..6, cols 0..2 -> broadcast from lanes 0..2.
    float ca[8];
    #pragma unroll
    for (int m=0;m<8;++m) ca[m] = acc[m];

    float num[3][3], den[3], G[3][3];
    #pragma unroll
    for (int k=0;k<3;++k){
      den[k] = __shfl(ca[3], k, 32);
      #pragma unroll
      for (int c=0;c<3;++c)  num[c][k] = __shfl(ca[c],   k, 32);
      #pragma unroll
      for (int kk=0;kk<3;++kk) G[kk][k] = __shfl(ca[4+kk], k, 32);
    }
    float col[3][3];
    #pragma unroll
    for (int k=0;k<3;++k){
      float dinv = 1.0f / (den[k] + 1e-10f);
      #pragma unroll
      for (int c=0;c<3;++c) col[c][k] = num[c][k] * dinv;
    }
    float t1 = 0.f, t2 = 0.f;
    #pragma unroll
    for (int c=0;c<3;++c){
      #pragma unroll
      for (int k=0;k<3;++k){
        t1 += col[c][k]*num[c][k];
        #pragma unroll
        for (int kp=0;kp<3;++kp) t2 += col[c][k]*col[c][kp]*G[k][kp];
      }
    }
    if (colOut){
      #pragma unroll
      for (int c=0;c<3;++c)
        #pragma unroll
        for (int k=0;k<3;++k) colOut[c*3+k] = col[c][k];
    }
    return img2 - 2.0f*t1 + t2;   // = SSE; monotone in reference mean loss
  };

  #pragma unroll 1
  for (int ip=0; ip<5; ++ip){
    float bestLoss = 3.4e38f;
    float bestR = 0.0f;
    #pragma unroll 1
    for (int v=0; v<NVC; ++v){
      // angle_range = linspace(0,2pi,32)[:31]; xy_range = linspace(-3,3,31)
      float rv = (ip < 3) ? (TWO_PI_F/31.0f)*(float)v : (-3.0f + 0.2f*(float)v);
      float q[5] = {prm[0],prm[1],prm[2],prm[3],prm[4]};
      q[ip] += rv;
      Geom g;
      float loss = evaluate(q, g, nullptr);
      if (loss < bestLoss){ bestLoss = loss; bestR = rv; }  // first-min tiebreak
    }
    prm[ip] += bestR;
  }

  (void)evaluate(prm, gfin, colfin);

  if (lane == 0){
    float* r = rec + (size_t)patch * RECN;
    r[0]=gfin.x0;   r[1]=gfin.y0;
    r[2]=gfin.s1;   r[3]=gfin.c1;
    r[4]=gfin.s2;   r[5]=gfin.c2;
    r[6]=gfin.s3;   r[7]=gfin.c3;
    r[8]=gfin.s4;   r[9]=gfin.c4;
    r[10]=gfin.sg13; r[11]=gfin.sg42;
    r[12]=gfin.g13;  r[13]=gfin.g42;
    #pragma unroll
    for (int t=0;t<9;++t) r[14+t] = colfin[t];
    r[23] = 0.f;
  }
}

// ---------------------------------------------------------------------------
// Kernel 2: gather-style fold. Each output pixel visits its <=4 contributing
// patches, recomputes the wedge values from the per-patch record, accumulates
// reconstructed color + boundary, and divides by the contribution count
// (== fold(ones)). Deterministic, no atomics; rec stays L2-resident (~380KB).
// ---------------------------------------------------------------------------
__global__ void foj_render(const float* __restrict__ rec, float* __restrict__ out){
  int idx = blockIdx.x * blockDim.x + threadIdx.x;
  if (idx >= HH*WW) return;
  int y  = idx >> 8;
  int xg = idx & 255;
  int hlo = (y  >= PS) ? ((y  - PS)/STR + 1) : 0;
  int hhi = min(HPN-1, y  / STR);
  int wlo = (xg >= PS) ? ((xg - PS)/STR + 1) : 0;
  int whi = min(WPN-1, xg / STR);

  float accC0=0.f, accC1=0.f, accC2=0.f, accB=0.f;
  int cnt = 0;
  for (int hp=hlo; hp<=hhi; ++hp){
    int   i  = y - hp*STR;
    float yv = -1.0f + (2.0f/7.0f)*(float)i;
    for (int wq=wlo; wq<=whi; ++wq){
      int   j  = xg - wq*STR;
      float xv = -1.0f + (2.0f/7.0f)*(float)j;
      const float* r = rec + (size_t)(hp*WPN + wq) * RECN;
      Geom g;
      g.x0=r[0];  g.y0=r[1];
      g.s1=r[2];  g.c1=r[3];  g.s2=r[4];  g.c2=r[5];
      g.s3=r[6];  g.c3=r[7];  g.s4=r[8];  g.c4=r[9];
      g.sg13=r[10]; g.sg42=r[11]; g.g13=r[12]; g.g42=r[13];
      float d13, d42;
      dists_at(g, xv, yv, d13, d42);
      float w0, w1, w2;
      wedges_from(d13, d42, w0, w1, w2);
      accC0 += w0*r[14] + w1*r[15] + w2*r[16];
      accC1 += w0*r[17] + w1*r[18] + w2*r[19];
      accC2 += w0*r[20] + w1*r[21] + w2*r[22];
      float minabs = (d13 < 0.f) ? -d13
                   : ((d42 < 0.f) ? fminf(d13, -d42) : fminf(d13, d42));
      float ratio = minabs * (1.0f/DELTA_F);
      accB += 1.0f / (1.0f + ratio*ratio);
      cnt++;
    }
  }
  float inv = 1.0f / (float)cnt;
  out[0*(HH*WW) + idx] = accC0 * inv;   // global_image, channel 0
  out[1*(HH*WW) + idx] = accC1 * inv;   // channel 1
  out[2*(HH*WW) + idx] = accC2 * inv;   // channel 2
  out[3*(HH*WW) + idx] = accB  * inv;   // global_boundaries
}

extern "C" void kernel_launch(void* const* d_in, const int* in_sizes, int n_in,
                              void* d_out, int out_size, void* d_ws, size_t ws_size,
                              hipStream_t stream){
  (void)in_sizes; (void)n_in; (void)out_size; (void)ws_size;
  const float* x  = (const float*)d_in[0];
  float* out = (float*)d_out;
  float* rec = (float*)d_ws;               // NPATCH * RECN floats (~381 KB)
  foj_descent<<<NPATCH, 32, 0, stream>>>(x, rec);
  foj_render<<<(HH*WW + 255)/256, 256, 0, stream>>>(rec, out);
}